// MiniRocketFeatureExtractor_9835475108405
// MI455X (gfx1250) — compile-verified
//
#include <hip/hip_runtime.h>

typedef __attribute__((ext_vector_type(2))) float v2f;
typedef __attribute__((ext_vector_type(8))) float v8f;

#define B_   32      // batch
#define L_   4096    // input length
#define C_   8       // channels
#define D_   6       // dilations (1<<d)
#define K_   84      // kernels
#define F_   4       // feats per (dil,kernel)
#define MT_  6       // ceil(84/16) M-tiles
#define NT_  (L_ / 16)  // 256 t-tiles

// ---------------------------------------------------------------------------
// Kernel A: negated dilated 9-tap box sum, conv layout [d][b][t][c]
// channel-fastest tid decode -> x reads AND conv stores are contiguous;
// conv[tid] == conv[((d*32+b)*4096+t)*8+c].
// ---------------------------------------------------------------------------
__global__ __launch_bounds__(256) void mr_conv(const float* __restrict__ x,
                                               float* __restrict__ conv) {
  unsigned tid = blockIdx.x * 256u + threadIdx.x;
  int c = tid & 7;
  int t = (tid >> 3) & (L_ - 1);
  int b = (tid >> 15) & 31;
  int d = tid >> 20;
  int dil = 1 << d;
  float s = 0.f;
#pragma unroll
  for (int w = 0; w < 9; ++w) {
    int tt = t + (w - 4) * dil;
    if (tt >= 0 && tt < L_) s += x[((b << 12) + tt) * C_ + c];
  }
  conv[tid] = -s;
}

// ---------------------------------------------------------------------------
// Kernel B: one wave per (b, d, mtile).
//   resp tile [16 kernels x 16 t] = masks(16x8) @ conv(8x16)
//   via two chained V_WMMA_F32_16X16X4_F32 (K = channels 0..3, then 4..7).
// A 16x4 layout: lanes 0-15 = M rows, VGPR0=K0 / VGPR1=K1; lanes16-31: K2/K3.
// B 4x16 layout: lanes 0-15 = N cols, VGPR0=K0 (half1: K2), VGPR1=K1 (K3).
// D 16x16 layout: VGPR i -> M=i (lanes 0-15) / M=i+8 (lanes 16-31), N=lane%16.
//
// PPV counting: cnt_full over all t everywhere; cnt_edge only in the <=16
// boundary tiles (pad <= 128).  cnt_trim = cnt_full - cnt_edge.  Interior
// loop is software-pipelined: B fragments for tile nt+1 are loaded before the
// WMMAs + compare/count of tile nt, so global latency hides behind VALU work.
// ---------------------------------------------------------------------------
__global__ __launch_bounds__(32) void mr_wmma_ppv(
    const float* __restrict__ conv, const float* __restrict__ masks,
    const float* __restrict__ biases, const float* __restrict__ fmean,
    const float* __restrict__ fstd, float* __restrict__ out) {
  int wid   = blockIdx.x;
  int mtile = wid % MT_;
  int d     = (wid / MT_) % D_;
  int b     = wid / (MT_ * D_);
  int lane  = threadIdx.x;      // 0..31 (wave32)
  int n     = lane & 15;        // N index / M row for A
  int khi   = lane >> 4;        // K-half select

  int dil = 1 << d;
  int pad = 4 * dil;            // 4..128; trim range [pad, L-pad) always valid

  // ---- A fragments (channel masks), two K-steps of 4 channels ----
  int ka   = mtile * 16 + n;
  float aw = (ka < K_) ? 1.f : 0.f;          // zero-pad rows 84..95
  int kam  = (ka < K_) ? ka : 0;
  const float* mrow = masks + (d * K_ + kam) * C_;
  int c0 = 2 * khi;                           // lane-half channel base
  v2f a0, a1;
  a0.x = mrow[c0 + 0] * aw;  a0.y = mrow[c0 + 1] * aw;   // channels 0..3
  a1.x = mrow[c0 + 4] * aw;  a1.y = mrow[c0 + 5] * aw;   // channels 4..7

  // ---- per-lane bias preload for the 8 D-fragment rows ----
  float bias_v[32];
#pragma unroll
  for (int i = 0; i < 8; ++i) {
    int kd  = mtile * 16 + i + 8 * khi;
    int kdm = (kd < K_) ? kd : (K_ - 1);     // clamp; garbage rows guarded later
#pragma unroll
    for (int f = 0; f < F_; ++f)
      bias_v[i * 4 + f] = biases[(d * K_ + kdm) * F_ + f];
  }

  float cnt_full[32], cnt_edge[32];
#pragma unroll
  for (int j = 0; j < 32; ++j) { cnt_full[j] = 0.f; cnt_edge[j] = 0.f; }

  // this lane's column within the [t][c] conv plane
  const float* cbl = conv + (((size_t)(d * B_ + b)) << 12) * C_ + (size_t)n * C_;

  auto loadB = [&](int nt, v2f& bf0, v2f& bf1) {
    const float* pt = cbl + (size_t)nt * (16 * C_);
    bf0.x = pt[c0 + 0];  bf0.y = pt[c0 + 1];   // 8B-aligned channel pairs
    bf1.x = pt[c0 + 4];  bf1.y = pt[c0 + 5];
  };
  auto mmacc = [&](v2f bf0, v2f bf1) -> v8f {
    v8f z = {};
    v8f acc = __builtin_amdgcn_wmma_f32_16x16x4_f32(
        false, a0, false, bf0, (short)0, z, false, false);
    return __builtin_amdgcn_wmma_f32_16x16x4_f32(
        false, a1, false, bf1, (short)0, acc, false, false);
  };

  auto edge_tile = [&](int nt) {
    v2f bf0, bf1;
    loadB(nt, bf0, bf1);
    v8f acc = mmacc(bf0, bf1);
    int t = nt * 16 + n;
    float edgew = (t < pad || t >= L_ - pad) ? 1.f : 0.f;
#pragma unroll
    for (int i = 0; i < 8; ++i) {
      float r = acc[i];
#pragma unroll
      for (int f = 0; f < 4; ++f) {
        bool gt = r > bias_v[i * 4 + f];
        cnt_full[i * 4 + f] += gt ? 1.f   : 0.f;
        cnt_edge[i * 4 + f] += gt ? edgew : 0.f;
      }
    }
  };

  int ilo = (pad + 15) >> 4;            // first fully-interior tile
  int ihi = (L_ - pad - 16) >> 4;       // last fully-interior tile

  for (int nt = 0; nt < ilo; ++nt) edge_tile(nt);          // head (<=8 tiles)

  // ---- interior: software-pipelined (load nt+1, compute nt) ----
  v2f pb0, pb1;
  loadB(ilo, pb0, pb1);
#pragma unroll 2
  for (int nt = ilo; nt <= ihi; ++nt) {
    v2f nb0, nb1;
    loadB(nt + 1, nb0, nb1);            // nt+1 <= ihi+1 <= 255: in bounds
    v8f acc = mmacc(pb0, pb1);
#pragma unroll
    for (int i = 0; i < 8; ++i) {
      float r = acc[i];
#pragma unroll
      for (int f = 0; f < 4; ++f)
        cnt_full[i * 4 + f] += (r > bias_v[i * 4 + f]) ? 1.f : 0.f;
    }
    pb0 = nb0;  pb1 = nb1;
  }

  for (int nt = ihi + 1; nt < NT_; ++nt) edge_tile(nt);    // tail (<=8 tiles)

  // ---- reduce the 16 t-phases within each lane half, then write ----
  float inv_full = 1.f / (float)L_;
  float inv_trim = 1.f / (float)(L_ - 2 * pad);
#pragma unroll
  for (int j = 0; j < 32; ++j) {
    float vF = cnt_full[j], vE = cnt_edge[j];
#pragma unroll
    for (int m = 1; m <= 8; m <<= 1) {
      vF += __shfl_xor(vF, m, 32);
      vE += __shfl_xor(vE, m, 32);
    }
    if (n == 0) {
      int i = j >> 2, f = j & 3;
      int k = mtile * 16 + i + 8 * khi;
      if (k < K_) {
        bool parity = ((d + k) & 1) == 1;          // odd -> trimmed mean
        float mean  = parity ? (vF - vE) * inv_trim : vF * inv_full;
        int fi = (d * K_ + k) * F_ + f;
        out[b * (D_ * K_ * F_) + fi] = (mean - fmean[fi]) / fstd[fi];
      }
    }
  }
}

// ---------------------------------------------------------------------------
extern "C" void kernel_launch(void* const* d_in, const int* in_sizes, int n_in,
                              void* d_out, int out_size, void* d_ws, size_t ws_size,
                              hipStream_t stream) {
  (void)in_sizes; (void)n_in; (void)out_size; (void)ws_size;
  const float* x      = (const float*)d_in[0];  // [32,4096,8]
  const float* masks  = (const float*)d_in[1];  // [6,84,8]
  const float* biases = (const float*)d_in[2];  // [6,84,4]
  const float* fmean  = (const float*)d_in[3];  // [2016]
  const float* fstd   = (const float*)d_in[4];  // [2016]
  float* out  = (float*)d_out;                  // [32,2016]
  float* conv = (float*)d_ws;                   // 6*32*4096*8 floats = 24 MB

  int totalA = D_ * B_ * C_ * L_;               // 6,291,456 (multiple of 256)
  mr_conv<<<totalA / 256, 256, 0, stream>>>(x, conv);
  mr_wmma_ppv<<<B_ * D_ * MT_, 32, 0, stream>>>(conv, masks, biases, fmean, fstd, out);
}